// SoftCLDiceLoss_48808008352163
// MI455X (gfx1250) — compile-verified
//
#include <hip/hip_runtime.h>
#include <hip/hip_bf16.h>
#include <stdint.h>

// ---------------------------------------------------------------------------
// soft-clDice for (16,1,1024,1024) f32 on MI455X (gfx1250, wave32).
// 21 skeletonization steps fused 3-per-launch in LDS (halo 4); skel held in
// registers across fused steps; delta phase uses vertical 16-row runs with a
// rolling separable 3x3 max (3.4 LDS reads/px instead of 9). CDNA5 path:
// global_load_async_to_lds_b128/b32 (ASYNCcnt) stages tiles straight to LDS.
// ---------------------------------------------------------------------------

#define IMG   1024
#define HW    (IMG * IMG)
#define NIMG  32                 // images 0..15 = "true", 16..31 = "pred"
#define TW    256                // tile width  (grid.x = 4)
#define TH    64                 // tile height (grid.y = 16)
#define KST   3                  // fused steps per launch (21 = 3*7)
#define HALO  (KST + 1)          // 4
#define PW    (TW + 2 * HALO)    // 264 (row = 1056 B, 16B-aligned chunks)
#define PH    (TH + 2 * HALO)    // 72
#define NTHR  1024
#define PIX_PER_THR 16           // TW*TH / NTHR

// CDNA5 async global->LDS copies (ASYNCcnt-tracked, no VGPR round-trip).
__device__ __forceinline__ void async_ld_f32(unsigned lds_off, const float* gp) {
    asm volatile("global_load_async_to_lds_b32 %0, %1, off"
                 :: "v"(lds_off), "v"(gp)
                 : "memory");
}
__device__ __forceinline__ void async_ld_b128(unsigned lds_off, const float* gp) {
    asm volatile("global_load_async_to_lds_b128 %0, %1, off"
                 :: "v"(lds_off), "v"(gp)
                 : "memory");
}
__device__ __forceinline__ void wait_async0() {
    asm volatile("s_wait_asynccnt 0" ::: "memory");
}

// Three fused steps of: E=erode5(x); D=dilate3x3(E); delta=relu(x-D);
// sk += relu(delta - sk*delta); x <- E.  Border clamp re-applied per step.
__global__ __launch_bounds__(NTHR) void clde_step3(
    const float* __restrict__ srcT,   // x for images 0..15
    const float* __restrict__ srcP,   // x for images 16..31
    float* __restrict__ Eout,         // next x (32-image buffer)
    float* __restrict__ skel,         // skeleton accumulator (32 images)
    int first, int last)
{
    __shared__ float planes[2][PH * PW];   // 2 x 76,032 B = 148.5 KB

    const int tid = threadIdx.x;
    const int img = blockIdx.z;
    const float* src = (img < 16) ? (srcT + (size_t)img * HW)
                                  : (srcP + (size_t)(img - 16) * HW);
    const int X0 = blockIdx.x * TW;
    const int Y0 = blockIdx.y * TH;
    const int topV = (blockIdx.y == 0);
    const int botV = (blockIdx.y == (IMG / TH - 1));
    const int lftV = (blockIdx.x == 0);
    const int rgtV = (blockIdx.x == (IMG / TW - 1));

    // Thread -> pixel mapping: column tx, contiguous vertical run of 16 rows.
    const int tx  = tid & (TW - 1);
    const int ty0 = (tid >> 8) << 4;          // 0,16,32,48

    // ---- Stage x tile (halo 4) into LDS via async DMA, clamp-to-edge. ----
    // 4-float chunks are 16B-aligned in both global and LDS; only the edge
    // chunks of image-border blocks need the element-wise clamped fallback.
    {
        const int NCH = PW / 4;               // 66 chunks per plane row
        for (int idx = tid; idx < PH * NCH; idx += NTHR) {
            int r  = idx / NCH;
            int cc = (idx - r * NCH) * 4;     // plane col of chunk start
            int gr = Y0 - HALO + r;
            gr = gr < 0 ? 0 : (gr > IMG - 1 ? IMG - 1 : gr);
            const float* grow = src + (size_t)gr * IMG;
            int gcol = X0 - HALO + cc;
            unsigned lds = (unsigned)(uintptr_t)&planes[0][r * PW + cc];
            if (gcol >= 0 && gcol + 3 <= IMG - 1) {
                async_ld_b128(lds, grow + gcol);
            } else {
                for (int e = 0; e < 4; ++e) {
                    int gc = gcol + e;
                    gc = gc < 0 ? 0 : (gc > IMG - 1 ? IMG - 1 : gc);
                    async_ld_f32(lds + 4u * e, grow + gc);
                }
            }
        }
    }

    // ---- Load skel into registers (overlaps the async DMA). ----
    float sk[PIX_PER_THR];
    if (first) {
        for (int k = 0; k < PIX_PER_THR; ++k) sk[k] = 0.0f;
    } else {
        for (int k = 0; k < PIX_PER_THR; ++k)
            sk[k] = skel[(size_t)img * HW +
                         (size_t)(Y0 + ty0 + k) * IMG + (X0 + tx)];
    }

    wait_async0();
    __syncthreads();

    int pp = 0;
    for (int j = 0; j < KST; ++j) {
        const float* S = planes[pp];        // x_j
        float* T = planes[pp ^ 1];          // E_j

        // E = min(up,down,left,right,center) over plane minus 1-ring.
        for (int idx = tid; idx < (PH - 2) * (PW - 2); idx += NTHR) {
            int r = idx / (PW - 2);         // 0..PH-3
            int c = idx - r * (PW - 2);     // 0..PW-3
            const float* s0 = &S[r * PW + (c + 1)];   // (r, c+1): above center
            float up = s0[0];
            float lf = s0[PW - 1];
            float ce = s0[PW];
            float rt = s0[PW + 1];
            float dn = s0[2 * PW];
            T[(r + 1) * PW + (c + 1)] =
                fminf(fminf(fminf(up, dn), fminf(lf, rt)), ce);
        }
        __syncthreads();

        // Re-apply clamp-to-edge (reference clamps at EVERY step): overwrite
        // out-of-image halo cells of T from the image-edge row/col. Uniform
        // per-block branches; no-op for interior blocks.
        if (topV) {
            for (int idx = tid; idx < HALO * PW; idx += NTHR) {
                int r = idx / PW, c = idx - r * PW;
                int cs = c;
                if (lftV && cs < HALO) cs = HALO;
                if (rgtV && cs > PW - 1 - HALO) cs = PW - 1 - HALO;
                T[r * PW + c] = T[HALO * PW + cs];
            }
        }
        if (botV) {
            for (int idx = tid; idx < HALO * PW; idx += NTHR) {
                int r = idx / PW, c = idx - r * PW;
                int cs = c;
                if (lftV && cs < HALO) cs = HALO;
                if (rgtV && cs > PW - 1 - HALO) cs = PW - 1 - HALO;
                T[(PH - HALO + r) * PW + c] = T[(PH - HALO - 1) * PW + cs];
            }
        }
        {
            const int rlo = topV ? HALO : 0;
            const int rhi = botV ? (PH - HALO - 1) : (PH - 1);
            const int nr  = rhi - rlo + 1;
            if (lftV) {
                for (int idx = tid; idx < HALO * nr; idx += NTHR) {
                    int r = rlo + idx / HALO, c = idx % HALO;
                    T[r * PW + c] = T[r * PW + HALO];
                }
            }
            if (rgtV) {
                for (int idx = tid; idx < HALO * nr; idx += NTHR) {
                    int r = rlo + idx / HALO, c = idx % HALO;
                    T[r * PW + (PW - HALO + c)] = T[r * PW + (PW - HALO - 1)];
                }
            }
        }
        __syncthreads();

        // delta + skel update: vertical run of 16 px/thread, rolling
        // separable 3x3 max (hm = horizontal max3 of one T row).
        {
            const int pc = tx + HALO;                 // 4..259
            const float* Tb = &T[pc];
            const float* Sb = &S[pc];
            const int r0 = ty0 + HALO - 1;            // first hm row (>=3)
            const float* t0 = &Tb[r0 * PW];
            float hm0 = fmaxf(fmaxf(t0[-1], t0[0]), t0[1]);
            const float* t1 = &Tb[(r0 + 1) * PW];
            float hm1 = fmaxf(fmaxf(t1[-1], t1[0]), t1[1]);
            for (int k = 0; k < PIX_PER_THR; ++k) {
                const float* t2 = &Tb[(r0 + 2 + k) * PW];
                float hm2 = fmaxf(fmaxf(t2[-1], t2[0]), t2[1]);
                float m  = fmaxf(fmaxf(hm0, hm1), hm2);
                float xv = Sb[(ty0 + HALO + k) * PW];
                float delta = fmaxf(xv - m, 0.0f);
                sk[k] = sk[k] + fmaxf(delta - sk[k] * delta, 0.0f);
                hm0 = hm1;
                hm1 = hm2;
            }
        }
        __syncthreads();   // before next step overwrites S
        pp ^= 1;
    }

    // ---- Write back skel and next-x (= E of last fused step). ----
    const float* Xn = planes[pp];
    for (int k = 0; k < PIX_PER_THR; ++k) {
        size_t gi = (size_t)img * HW +
                    (size_t)(Y0 + ty0 + k) * IMG + (X0 + tx);
        skel[gi] = sk[k];
        if (!last) Eout[gi] = Xn[(ty0 + k + HALO) * PW + (tx + HALO)];
    }
}

__global__ void clde_zero4(float* sums) {
    if (threadIdx.x < 4) sums[threadIdx.x] = 0.0f;
}

// sums[0]=sum(skel_true) sums[1]=sum(skel_true*y_pred)
// sums[2]=sum(skel_pred) sums[3]=sum(skel_pred*y_true)
__global__ __launch_bounds__(256) void clde_reduce(
    const float* __restrict__ skel,
    const float* __restrict__ y_true,
    const float* __restrict__ y_pred,
    float* __restrict__ sums)
{
    const size_t Nh = (size_t)16 * HW;
    float s_t = 0.0f, stp = 0.0f, s_p = 0.0f, spt = 0.0f;
    size_t stride = (size_t)gridDim.x * 256;
    for (size_t i = (size_t)blockIdx.x * 256 + threadIdx.x; i < Nh; i += stride) {
        float a = skel[i];
        float b = skel[Nh + i];
        s_t += a;
        stp += a * y_pred[i];
        s_p += b;
        spt += b * y_true[i];
    }
    for (int o = 16; o > 0; o >>= 1) {
        s_t += __shfl_xor(s_t, o);
        stp += __shfl_xor(stp, o);
        s_p += __shfl_xor(s_p, o);
        spt += __shfl_xor(spt, o);
    }
    __shared__ float part[8][4];
    int wave = threadIdx.x >> 5;
    int lane = threadIdx.x & 31;
    if (lane == 0) {
        part[wave][0] = s_t; part[wave][1] = stp;
        part[wave][2] = s_p; part[wave][3] = spt;
    }
    __syncthreads();
    if (threadIdx.x < 4) {
        float t = 0.0f;
        for (int w = 0; w < 8; ++w) t += part[w][threadIdx.x];
        atomicAdd(&sums[threadIdx.x], t);
    }
}

__global__ void clde_final(const float* __restrict__ sums, float* __restrict__ out) {
    float s_true = sums[0], stp = sums[1], s_pred = sums[2], spt = sums[3];
    float tprec = (spt + 1.0f) / (s_pred + 1.0f);
    float tsens = (stp + 1.0f) / (s_true + 1.0f);
    out[0] = 1.0f - 2.0f * (tprec * tsens) / (tprec + tsens + 1e-8f);
}

extern "C" void kernel_launch(void* const* d_in, const int* in_sizes, int n_in,
                              void* d_out, int out_size, void* d_ws, size_t ws_size,
                              hipStream_t stream) {
    (void)in_sizes; (void)n_in; (void)out_size; (void)ws_size;
    const float* y_true = (const float*)d_in[0];
    const float* y_pred = (const float*)d_in[1];
    float* out = (float*)d_out;
    float* ws  = (float*)d_ws;

    const size_t N1 = (size_t)NIMG * HW;      // 33,554,432 floats (128 MB)
    float* bufA = ws;
    float* bufB = ws + N1;
    float* skel = ws + 2 * N1;
    float* sums = ws + 3 * N1;

    dim3 blk(NTHR, 1, 1);
    dim3 grd(IMG / TW, IMG / TH, NIMG);       // 4 x 16 x 32 = 2048 blocks

    // Launch 0: x = inputs, skel starts at 0 (steps 0..2).
    clde_step3<<<grd, blk, 0, stream>>>(y_true, y_pred, bufA, skel, 1, 0);

    // Launches 1..6: 3 steps each (steps 3..20), x ping-pongs A<->B.
    float* cur = bufA;
    float* nxt = bufB;
    for (int l = 1; l < 7; ++l) {
        clde_step3<<<grd, blk, 0, stream>>>(cur, cur + (size_t)16 * HW, nxt, skel,
                                            0, (l == 6) ? 1 : 0);
        float* t = cur; cur = nxt; nxt = t;
    }

    clde_zero4<<<1, 32, 0, stream>>>(sums);
    clde_reduce<<<2048, 256, 0, stream>>>(skel, y_true, y_pred, sums);
    clde_final<<<1, 1, 0, stream>>>(sums, out);
}